// Clustering_7687991460039
// MI455X (gfx1250) — compile-verified
//
#include <hip/hip_runtime.h>
#include <math.h>

typedef __attribute__((ext_vector_type(2))) float v2f;
typedef __attribute__((ext_vector_type(8))) float v8f;

#define B_    36
#define H_    8
#define L_    1152
#define DK_   72
#define DM_   576
#define LK_   128
#define B2_   18
#define C_    3
#define DKP_  80      // dk padded to multiple of 16
#define NROWS (B_*LK_*B2_)   // 82944

// ---------------- workspace layout (floats) ----------------
#define OFF_KP     0                         // B*H*LK*DK      = 2654208
#define OFF_VP     (OFF_KP + B_*H_*LK_*DK_)  // B*H*LK*DKP     = 2949120
#define OFF_CC     (OFF_VP + B_*H_*LK_*DKP_) // B*H*LK*DKP     = 2949120
#define OFF_CQ     (OFF_CC + B_*H_*LK_*DKP_) // B*LK*B2*C      = 248832
#define OFF_CK     (OFF_CQ + B_*LK_*B2_*C_)
#define OFF_MU     (OFF_CK + B_*LK_*B2_*C_)  // B*LK*B2        = 82944
#define OFF_CCF    (OFF_MU + B_*LK_*B2_)     // B*LK*B2*64     = 5308416
#define OFF_ACC    (OFF_CCF + B_*LK_*B2_*64)

// ---------------- init: zero Vp (pad cols) and loss accumulators ----------
__global__ void k_init(float* vp, float* accum, int n) {
  int i = blockIdx.x * blockDim.x + threadIdx.x;
  if (i < n) vp[i] = 0.f;
  if (i == 0) { accum[0] = 0.f; accum[1] = 0.f; }
}

// ---------------- maxpool (k=9, s=9, pad=4) on raw (b, 576, 1152) view ----
__global__ void k_pool(const float* __restrict__ K, const float* __restrict__ V,
                       float* __restrict__ Kp, float* __restrict__ Vp) {
  int i = blockIdx.x * blockDim.x + threadIdx.x;   // over B*DM*LK
  const int total = B_ * DM_ * LK_;
  if (i >= total) return;
  int col = i % LK_;  int t = i / LK_;
  int r   = t % DM_;  int bb = t / DM_;
  const float* src = (blockIdx.y == 0) ? K : V;
  const float* row = src + (size_t)bb * DM_ * L_ + (size_t)r * L_;
  int lo = col * 9 - 4, hi = lo + 9;
  if (lo < 0) lo = 0;
  if (hi > L_) hi = L_;
  float mx = -INFINITY;
  for (int j = lo; j < hi; ++j) mx = fmaxf(mx, row[j]);
  // raw reshape (576,128) -> (h, lk, dk)
  int flat = r * LK_ + col;
  int h  = flat / (LK_ * DK_);
  int rem = flat % (LK_ * DK_);
  int lk = rem / DK_;
  int dk = rem % DK_;
  if (blockIdx.y == 0)
    Kp[(((bb * H_ + h) * LK_ + lk) * DK_) + dk] = mx;
  else
    Vp[(((bb * H_ + h) * LK_ + lk) * DKP_) + dk] = mx;
}

__device__ inline void softmax3(float* s) {
  float mx = fmaxf(s[0], fmaxf(s[1], s[2]));
  float e0 = expf(s[0] - mx), e1 = expf(s[1] - mx), e2 = expf(s[2] - mx);
  float inv = 1.f / (e0 + e1 + e2);
  s[0] = e0 * inv; s[1] = e1 * inv; s[2] = e2 * inv;
}

// ---------------- cluster path: one wave per (bt, lk, p) row -------------
__global__ void k_cluster(const float* __restrict__ Kp,
                          const float* __restrict__ Wp, const float* __restrict__ bp,
                          const float* __restrict__ Wck, const float* __restrict__ bck,
                          const float* __restrict__ Wcq, const float* __restrict__ bcq,
                          float* __restrict__ cq, float* __restrict__ ck,
                          float* __restrict__ mu, float* accum) {
  int wave = (int)((blockIdx.x * blockDim.x + threadIdx.x) >> 5);
  int lane = threadIdx.x & 31;
  if (wave >= NROWS) return;
  int p_o  = wave % B2_;  int t = wave / B2_;
  int lk_o = t % LK_;     int bt = t / LK_;
  float a0 = 0.f, a1 = 0.f, a2 = 0.f;
  for (int d = lane; d < DM_; d += 32) {
    // unfold/transpose/reshape index algebra:
    // f = ((h*128+lk)*72+dk)*18+pw  ==  lk_o*10368 + p_o*576 + d
    int f  = lk_o * (B2_ * DM_) + p_o * DM_ + d;
    int pw = f % B2_;  int u  = f / B2_;
    int dk = u % DK_;  int u2 = u / DK_;
    int lk = u2 % LK_; int h  = u2 / LK_;
    int src = bt + pw;                 // index into K_padded
    float val = 0.f;
    if (src >= B2_)                    // K_padded[j] = Kp[j-17] for j>=18
      val = Kp[(((src - 17) * H_ + h) * LK_ + lk) * DK_ + dk];
    a0 += val * Wp[d * C_ + 0];
    a1 += val * Wp[d * C_ + 1];
    a2 += val * Wp[d * C_ + 2];
  }
  for (int off = 16; off; off >>= 1) {
    a0 += __shfl_xor(a0, off, 32);
    a1 += __shfl_xor(a1, off, 32);
    a2 += __shfl_xor(a2, off, 32);
  }
  if (lane == 0) {
    float kp0 = fmaxf(a0 + bp[0], 0.f);
    float kp1 = fmaxf(a1 + bp[1], 0.f);
    float kp2 = fmaxf(a2 + bp[2], 0.f);
    float sk[3], sq[3];
    for (int j = 0; j < 3; ++j) {
      sk[j] = kp0 * Wck[0 * 3 + j] + kp1 * Wck[1 * 3 + j] + kp2 * Wck[2 * 3 + j] + bck[j];
      sq[j] = kp0 * Wcq[0 * 3 + j] + kp1 * Wcq[1 * 3 + j] + kp2 * Wcq[2 * 3 + j] + bcq[j];
    }
    softmax3(sk);
    softmax3(sq);
    float mu_ = (sq[0] + sq[1] + sq[2]) * (1.f / 3.f);
    float d0 = sq[0] - mu_, d1 = sq[1] - mu_, d2 = sq[2] - mu_;
    float sd = sqrtf((d0 * d0 + d1 * d1 + d2 * d2) * 0.5f);   // unbiased std, n=3
    float sigma = (sd > 20.f) ? sd : log1pf(expf(sd));        // softplus
    float xk = (sk[0] + sk[1] + sk[2]) * (1.f / 3.f);
    float z = (xk - mu_) / sigma;
    float lp = -0.5f * z * z - logf(sigma) - 0.91893853320467274f;
    atomicAdd(&accum[0], lp);
    int base = wave * C_;
    for (int j = 0; j < 3; ++j) { cq[base + j] = sq[j]; ck[base + j] = sk[j]; }
    mu[wave] = mu_;
  }
}

// ---------------- CE loss: softmax of mu over the l_k axis ---------------
__global__ void k_ce(const float* __restrict__ mu, float* accum) {
  __shared__ float red[LK_];
  int b = blockIdx.x / B2_, p = blockIdx.x % B2_;
  int lk = threadIdx.x;
  float v = mu[(b * LK_ + lk) * B2_ + p];
  red[lk] = v; __syncthreads();
  for (int s = 64; s; s >>= 1) { if (lk < s) red[lk] = fmaxf(red[lk], red[lk + s]); __syncthreads(); }
  float mx = red[0]; __syncthreads();
  red[lk] = expf(v - mx); __syncthreads();
  for (int s = 64; s; s >>= 1) { if (lk < s) red[lk] += red[lk + s]; __syncthreads(); }
  float lse = mx + logf(red[0]); __syncthreads();
  red[lk] = v * (v - lse); __syncthreads();
  for (int s = 64; s; s >>= 1) { if (lk < s) red[lk] += red[lk + s]; __syncthreads(); }
  if (lk == 0) atomicAdd(&accum[1], -red[0]);
}

// ---------------- 18x18 causal mini-attention + Wb + pool ----------------
__global__ void k_scores(const float* __restrict__ cq, const float* __restrict__ ck,
                         const float* __restrict__ Wb, const float* __restrict__ bb,
                         float* __restrict__ CCflat) {
  __shared__ float scq[B2_][C_], sck[B2_][C_], scq2[B2_][C_];
  int blk = blockIdx.x;                  // b*LK + lk
  int base = blk * B2_ * C_;
  int tid = threadIdx.x;
  if (tid < B2_ * C_) { ((float*)scq)[tid] = cq[base + tid]; ((float*)sck)[tid] = ck[base + tid]; }
  __syncthreads();
  if (tid < B2_) {
    int p = tid;
    float s[B2_];
    float mx = -INFINITY;
    for (int u = 0; u <= p; ++u) {       // u>p is masked to -1e9 -> exp==0
      float d = (scq[p][0] * sck[u][0] + scq[p][1] * sck[u][1] + scq[p][2] * sck[u][2]) * (1.f / 3.f);
      s[u] = d; mx = fmaxf(mx, d);
    }
    float sum = 0.f;
    for (int u = 0; u <= p; ++u) { s[u] = expf(s[u] - mx); sum += s[u]; }
    float inv = 1.f / sum;
    float o0 = 0.f, o1 = 0.f, o2 = 0.f;
    for (int u = 0; u <= p; ++u) {
      float w = s[u] * inv;
      o0 += w * scq[u][0]; o1 += w * scq[u][1]; o2 += w * scq[u][2];
    }
    scq2[p][0] = o0; scq2[p][1] = o1; scq2[p][2] = o2;
  }
  __syncthreads();
  for (int out = tid; out < B2_ * 64; out += blockDim.x) {
    int p = out >> 6, j = out & 63;
    int lo = j * 9 - 4, hi = lo + 9;
    if (lo < 0) lo = 0;
    if (hi > DM_) hi = DM_;
    float c0 = scq2[p][0], c1 = scq2[p][1], c2 = scq2[p][2];
    float mx = -INFINITY;
    for (int d = lo; d < hi; ++d) {
      float v = c0 * Wb[d] + c1 * Wb[DM_ + d] + c2 * Wb[2 * DM_ + d] + bb[d];
      mx = fmaxf(mx, v);
    }
    CCflat[(size_t)blk * (B2_ * 64) + out] = fmaxf(mx, 0.f);  // relu(max)==max(relu)
  }
}

// ---------------- sum over c=2 of reshaped cluster_center ----------------
__global__ void k_ccsum(const float* __restrict__ CCflat, float* __restrict__ CC) {
  int i = blockIdx.x * blockDim.x + threadIdx.x;
  const int total = B_ * H_ * LK_ * DKP_;
  if (i >= total) return;
  int d = i % DKP_;  int t  = i / DKP_;
  int k = t % LK_;   int t2 = t / LK_;
  int h = t2 % H_;   int b  = t2 / H_;
  float v = 0.f;
  if (d < DK_) {
    const float* base = CCflat + (size_t)b * (LK_ * B2_ * 64);
    v = base[((h * 2 + 0) * LK_ + k) * DK_ + d] +
        base[((h * 2 + 1) * LK_ + k) * DK_ + d];
  }
  CC[i] = v;
}

// ---------------- fused flash attention with f32 WMMA --------------------
// One wave per 16-row Q tile: S(16x128)=Q@CCsum^T via v_wmma_f32_16x16x4_f32,
// row softmax in LDS, O(16x72)=P@Vp via WMMA again.
__global__ __launch_bounds__(128) void k_attn(const float* __restrict__ Q,
                                              const float* __restrict__ CC,
                                              const float* __restrict__ Vp,
                                              float* __restrict__ Out) {
  __shared__ float sS[4][16][LK_];       // 32 KB: per-wave score tile
  int wv = threadIdx.x >> 5, lane = threadIdx.x & 31;
  int bh = blockIdx.y;
  int m0 = blockIdx.x * 64 + wv * 16;
  const float* Qb  = Q  + (size_t)bh * L_ * DK_;
  const float* CCb = CC + (size_t)bh * LK_ * DKP_;
  const float* Vb  = Vp + (size_t)bh * LK_ * DKP_;
  int ln = lane & 15;                    // M index (A) / N index (B)
  int kh = (lane >> 4) << 1;             // K pair base: lanes 16-31 hold K=2,3
  int mh = (lane >> 4) << 3;             // C/D rows: upper lanes hold M+8

  // ---- phase 1: S = Q @ CCsum^T ----
  v8f acc[8] = {};
  for (int k0 = 0; k0 < DK_; k0 += 4) {
    v2f a;
    a.x = Qb[(m0 + ln) * DK_ + k0 + kh];
    a.y = Qb[(m0 + ln) * DK_ + k0 + kh + 1];
    for (int nt = 0; nt < 8; ++nt) {
      v2f bf;                            // B[k][n] = CCsum[key=n][k]
      bf.x = CCb[(nt * 16 + ln) * DKP_ + k0 + kh];
      bf.y = CCb[(nt * 16 + ln) * DKP_ + k0 + kh + 1];
      acc[nt] = __builtin_amdgcn_wmma_f32_16x16x4_f32(
          false, a, false, bf, (short)0, acc[nt], false, false);
    }
  }
  for (int nt = 0; nt < 8; ++nt)
    for (int j = 0; j < 8; ++j)
      sS[wv][j + mh][nt * 16 + ln] = acc[nt][j];
  __syncthreads();

  // ---- row softmax over 128 keys ----
  if (lane < 16) {
    float* row = sS[wv][lane];
    float mx = -INFINITY;
    for (int n = 0; n < LK_; ++n) mx = fmaxf(mx, row[n]);
    float sum = 0.f;
    for (int n = 0; n < LK_; ++n) { float e = expf(row[n] - mx); row[n] = e; sum += e; }
    float inv = 1.f / sum;
    for (int n = 0; n < LK_; ++n) row[n] *= inv;
  }
  __syncthreads();

  // ---- phase 2: O = P @ Vp ----
  v8f acc2[5] = {};
  for (int k0 = 0; k0 < LK_; k0 += 4) {
    v2f a;
    a.x = sS[wv][ln][k0 + kh];
    a.y = sS[wv][ln][k0 + kh + 1];
    for (int nt = 0; nt < 5; ++nt) {
      v2f bf;                            // B[k][n] = Vp[key=k][d=n]
      bf.x = Vb[(k0 + kh) * DKP_ + nt * 16 + ln];
      bf.y = Vb[(k0 + kh + 1) * DKP_ + nt * 16 + ln];
      acc2[nt] = __builtin_amdgcn_wmma_f32_16x16x4_f32(
          false, a, false, bf, (short)0, acc2[nt], false, false);
    }
  }
  for (int nt = 0; nt < 5; ++nt) {
    int n = nt * 16 + ln;
    if (n < DK_)
      for (int j = 0; j < 8; ++j)
        Out[(size_t)bh * L_ * DK_ + (size_t)(m0 + j + mh) * DK_ + n] = acc2[nt][j];
  }
}

// ---------------- combine loss terms -------------------------------------
__global__ void k_loss(const float* accum, float* out) {
  out[0] = -(accum[0] / (float)NROWS) + (accum[1] / (float)(B_ * B2_));
}

extern "C" void kernel_launch(void* const* d_in, const int* in_sizes, int n_in,
                              void* d_out, int out_size, void* d_ws, size_t ws_size,
                              hipStream_t stream) {
  const float* Q   = (const float*)d_in[0];
  const float* K   = (const float*)d_in[1];
  const float* V   = (const float*)d_in[2];
  const float* Wp  = (const float*)d_in[3];
  const float* bp  = (const float*)d_in[4];
  const float* Wck = (const float*)d_in[5];
  const float* bck = (const float*)d_in[6];
  const float* Wcq = (const float*)d_in[7];
  const float* bcq = (const float*)d_in[8];
  const float* Wb  = (const float*)d_in[9];
  const float* bb  = (const float*)d_in[10];

  float* ws     = (float*)d_ws;
  float* Kp     = ws + OFF_KP;
  float* Vp     = ws + OFF_VP;
  float* CC     = ws + OFF_CC;
  float* cq     = ws + OFF_CQ;
  float* ck     = ws + OFF_CK;
  float* mu     = ws + OFF_MU;
  float* CCflat = ws + OFF_CCF;
  float* accum  = ws + OFF_ACC;

  float* out = (float*)d_out;

  // 1) zero Vp (for dk pad columns) + loss accumulators
  {
    int n = B_ * H_ * LK_ * DKP_;
    k_init<<<(n + 255) / 256, 256, 0, stream>>>(Vp, accum, n);
  }
  // 2) maxpool K -> Kp, V -> Vp(padded)
  {
    int n = B_ * DM_ * LK_;
    dim3 g((n + 255) / 256, 2);
    k_pool<<<g, 256, 0, stream>>>(K, V, Kp, Vp);
  }
  // 3) cluster path: one wave per row
  k_cluster<<<NROWS * 32 / 128, 128, 0, stream>>>(Kp, Wp, bp, Wck, bck, Wcq, bcq,
                                                  cq, ck, mu, accum);
  // 4) CE loss over l_k axis
  k_ce<<<B_ * B2_, LK_, 0, stream>>>(mu, accum);
  // 5) mini-attention + cluster center + pool
  k_scores<<<B_ * LK_, 64, 0, stream>>>(cq, ck, Wb, bb, CCflat);
  // 6) sum over c=2 into padded CCsum
  {
    int n = B_ * H_ * LK_ * DKP_;
    k_ccsum<<<(n + 255) / 256, 256, 0, stream>>>(CCflat, CC);
  }
  // 7) fused flash attention (WMMA f32)
  {
    dim3 g(L_ / 64, B_ * H_);
    k_attn<<<g, 128, 0, stream>>>(Q, CC, Vp, out);
  }
  // 8) loss scalar
  k_loss<<<1, 1, 0, stream>>>(accum, out + (size_t)B_ * H_ * L_ * DK_);
}